// Map_BWCaster_69260642615890
// MI455X (gfx1250) — compile-verified
//
#include <hip/hip_runtime.h>
#include <hip/hip_bf16.h>
#include <stdint.h>

// Problem constants (from setup_inputs): n_rays=2048, n_samp=256, G=128, S=6, C=16
#define GDIM 128
#define S_DIM 6
#define C_DIM 16
#define SC 96                       // S*C channels per plane
#define PLANE_STRIDE (GDIM * GDIM)  // floats per channel image
#define LINES_ELEMS (3 * SC * GDIM) // 36864 floats = 144 KB (fits 320KB/WGP LDS)
#define BLOCK 256

// CDNA5 async global->LDS path (gfx1250). Builtin signature per hipcc diag:
// void(int4 AS1*, int4 AS3*, imm offset, imm cpol)
#if defined(__gfx1250__) && __has_builtin(__builtin_amdgcn_global_load_async_to_lds_b128)
#define USE_ASYNC_LDS 1
#else
#define USE_ASYNC_LDS 0
#endif

typedef int v4i __attribute__((ext_vector_type(4)));

#if USE_ASYNC_LDS
__device__ __forceinline__ void async_copy16(const float* g, float* l) {
    // addrspacecast first (same pointee), then reinterpret to int4* in-AS.
    const __attribute__((address_space(1))) float* g1 =
        (const __attribute__((address_space(1))) float*)g;
    __attribute__((address_space(3))) float* l1 =
        (__attribute__((address_space(3))) float*)l;
    __builtin_amdgcn_global_load_async_to_lds_b128(
        (__attribute__((address_space(1))) v4i*)g1,
        (__attribute__((address_space(3))) v4i*)l1,
        0, 0);
}
#endif

__global__ __launch_bounds__(BLOCK) void warp_bwfield_kernel(
    const float* __restrict__ xyz,     // (N,3)
    const float* __restrict__ vdirs,   // (N,3)
    const float* __restrict__ planes,  // (3,96,128,128)
    const float* __restrict__ lines,   // (3,96,128)
    const float* __restrict__ aabb,    // (2,3)
    float* __restrict__ out,           // xw (N,3) then vw (N,3)
    int N)
{
    __shared__ float s_lines[LINES_ELEMS];

    // ---- Stage the full lines tensor into LDS (CDNA5 async copy) ----
#if USE_ASYNC_LDS
    {
        // 36864 floats = 9216 x 16B chunks; 256 threads x 36 chunks each.
        #pragma unroll
        for (int k = 0; k < LINES_ELEMS / (4 * BLOCK); ++k) {
            int chunk = threadIdx.x + k * BLOCK;
            async_copy16(lines + chunk * 4, &s_lines[chunk * 4]);
        }
#if __has_builtin(__builtin_amdgcn_s_wait_asynccnt)
        __builtin_amdgcn_s_wait_asynccnt(0);
#else
        asm volatile("s_wait_asynccnt 0" ::: "memory");
#endif
    }
#else
    for (int i = threadIdx.x; i < LINES_ELEMS; i += BLOCK) s_lines[i] = lines[i];
#endif
    __syncthreads();

    int n = blockIdx.x * blockDim.x + threadIdx.x;
    if (n >= N) return;

    // ---- Normalize coordinates ----
    float pt[3], dir[3], xn[3];
    #pragma unroll
    for (int d = 0; d < 3; ++d) {
        pt[d]  = xyz[3 * n + d];
        dir[d] = vdirs[3 * n + d];
        float inv = 2.0f / (aabb[3 + d] - aabb[d]);
        xn[d] = (pt[d] - aabb[d]) * inv - 1.0f;
    }

    // ---- Tri-plane / tri-line sampling, contract over C ----
    const int M0[3] = {0, 0, 1};   // MAT_MODE first coord  -> x (width)
    const int M1[3] = {1, 2, 2};   // MAT_MODE second coord -> y (height)
    const int VM[3] = {2, 1, 0};   // VEC_MODE

    float bwf[6] = {0.f, 0.f, 0.f, 0.f, 0.f, 0.f};

    #pragma unroll
    for (int p = 0; p < 3; ++p) {
        const float gm1 = (float)(GDIM - 1);
        float x = fminf(fmaxf((xn[M0[p]] + 1.0f) * 0.5f * gm1, 0.0f), gm1);
        float y = fminf(fmaxf((xn[M1[p]] + 1.0f) * 0.5f * gm1, 0.0f), gm1);
        float z = fminf(fmaxf((xn[VM[p]] + 1.0f) * 0.5f * gm1, 0.0f), gm1);

        int x0 = (int)x, y0 = (int)y, z0 = (int)z;
        int x1 = min(x0 + 1, GDIM - 1);
        int y1 = min(y0 + 1, GDIM - 1);
        int z1 = min(z0 + 1, GDIM - 1);
        float wx = x - (float)x0, wy = y - (float)y0, wz = z - (float)z0;
        float omx = 1.0f - wx, omy = 1.0f - wy, omz = 1.0f - wz;

        int o00 = y0 * GDIM + x0, o01 = y0 * GDIM + x1;
        int o10 = y1 * GDIM + x0, o11 = y1 * GDIM + x1;

        const float* pb = planes + (size_t)p * SC * PLANE_STRIDE;
        const float* lb = s_lines + p * SC * GDIM;

        #pragma unroll
        for (int s = 0; s < S_DIM; ++s) {
            float acc = 0.0f;
            #pragma unroll 4
            for (int c = 0; c < C_DIM; ++c) {
                const float* ch = pb + (size_t)(s * C_DIM + c) * PLANE_STRIDE;
                float f00 = ch[o00];
                float f01 = ch[o01];
                float f10 = ch[o10];
                float f11 = ch[o11];
                float top = f00 * omx + f01 * wx;
                float bot = f10 * omx + f11 * wx;
                float ps  = top * omy + bot * wy;

                const float* lc = lb + (s * C_DIM + c) * GDIM;
                float ls = lc[z0] * omz + lc[z1] * wz;

                acc += ps * ls;
            }
            bwf[s] += acc;
        }
    }

    // ---- SE(3) exponential map (Rodrigues, closed form) ----
    float w0 = bwf[0], w1 = bwf[1], w2 = bwf[2];
    float v0 = bwf[3], v1 = bwf[4], v2 = bwf[5];

    float tt = fmaxf(w0 * w0 + w1 * w1 + w2 * w2, 1e-6f);
    float theta = sqrtf(tt);
    float it = 1.0f / theta;
    w0 *= it; w1 *= it; w2 *= it;
    v0 *= it; v1 *= it; v2 *= it;

    float sn = sinf(theta), cs = cosf(theta);
    float omc = 1.0f - cs;   // 1 - cos(theta)
    float tms = theta - sn;  // theta - sin(theta)

    // R*x = cs*x + sn*(w x x) + omc*w*(w.x)
    float wdp = w0 * pt[0] + w1 * pt[1] + w2 * pt[2];
    float cp0 = w1 * pt[2] - w2 * pt[1];
    float cp1 = w2 * pt[0] - w0 * pt[2];
    float cp2 = w0 * pt[1] - w1 * pt[0];
    float rx0 = cs * pt[0] + sn * cp0 + omc * w0 * wdp;
    float rx1 = cs * pt[1] + sn * cp1 + omc * w1 * wdp;
    float rx2 = cs * pt[2] + sn * cp2 + omc * w2 * wdp;

    // t = P*v = sn*v + omc*(w x v) + (theta - sn)*w*(w.v)
    float wdv = w0 * v0 + w1 * v1 + w2 * v2;
    float cv0 = w1 * v2 - w2 * v1;
    float cv1 = w2 * v0 - w0 * v2;
    float cv2 = w0 * v1 - w1 * v0;
    float t0 = sn * v0 + omc * cv0 + tms * w0 * wdv;
    float t1 = sn * v1 + omc * cv1 + tms * w1 * wdv;
    float t2 = sn * v2 + omc * cv2 + tms * w2 * wdv;

    // vw = R*dir
    float wdd = w0 * dir[0] + w1 * dir[1] + w2 * dir[2];
    float cd0 = w1 * dir[2] - w2 * dir[1];
    float cd1 = w2 * dir[0] - w0 * dir[2];
    float cd2 = w0 * dir[1] - w1 * dir[0];
    float vw0 = cs * dir[0] + sn * cd0 + omc * w0 * wdd;
    float vw1 = cs * dir[1] + sn * cd1 + omc * w1 * wdd;
    float vw2 = cs * dir[2] + sn * cd2 + omc * w2 * wdd;

    // ---- Store: xw first, then vw ----
    out[3 * n + 0] = rx0 + t0;
    out[3 * n + 1] = rx1 + t1;
    out[3 * n + 2] = rx2 + t2;
    size_t voff = (size_t)N * 3;
    out[voff + 3 * n + 0] = vw0;
    out[voff + 3 * n + 1] = vw1;
    out[voff + 3 * n + 2] = vw2;
}

extern "C" void kernel_launch(void* const* d_in, const int* in_sizes, int n_in,
                              void* d_out, int out_size, void* d_ws, size_t ws_size,
                              hipStream_t stream) {
    const float* xyz    = (const float*)d_in[0];
    const float* vdirs  = (const float*)d_in[1];
    // d_in[2] = transforms (unused), d_in[3] = ray_valid (unused)
    const float* planes = (const float*)d_in[4];
    const float* lines  = (const float*)d_in[5];
    const float* aabb   = (const float*)d_in[6];
    float* out = (float*)d_out;

    int N = in_sizes[0] / 3;  // 2048*256 points
    dim3 grid((N + BLOCK - 1) / BLOCK);
    warp_bwfield_kernel<<<grid, BLOCK, 0, stream>>>(xyz, vdirs, planes, lines, aabb, out, N);
}